// CrossModalMamba_52664888984002
// MI455X (gfx1250) — compile-verified
//
#include <hip/hip_runtime.h>
#include <hip/hip_bf16.h>

// ---------------------------------------------------------------------------
// CrossModalMamba for MI455X (gfx1250, wave32).
// GEMMs via V_WMMA_F32_16X16X4_F32, LDS double-buffered tiling with
// gfx1250 async VMEM->LDS staging (GLOBAL_LOAD_ASYNC_TO_LDS_B128, ASYNCcnt).
// Selective scan: serial in L, parallel over B*D channels, LDS-broadcast B/C.
// ---------------------------------------------------------------------------

typedef __attribute__((ext_vector_type(2))) float v2f;
typedef __attribute__((ext_vector_type(8))) float v8f;

static constexpr int kB  = 4;
static constexpr int kL  = 1024;   // LT == LI
static constexpr int kD  = 512;
static constexpr int kN  = 16;     // d_state

// Async 16B copy global -> LDS (per lane), tracked by ASYNCcnt.
// LDS operand is the low 32 bits of the generic pointer (== LDS byte address).
__device__ __forceinline__ void async_b128(const float* g, float* l) {
  const unsigned lds_off = (unsigned)(size_t)l;
  asm volatile("global_load_async_to_lds_b128 %0, %1, off"
               :: "v"(lds_off), "v"(g) : "memory");
}
__device__ __forceinline__ void wait_async0() {
  asm volatile("s_wait_asynccnt 0x0" ::: "memory");
}

// ---------------------------------------------------------------------------
// Tiled WMMA GEMM: C[M,N] = A[M,K] @ B[K,N] (+bias) (+= existing C if ACCUM)
//   Block: 256 threads = 8 waves. Block tile BM=128 x BN=64, K-tile KT=32.
//   Wave w computes rows [w*16, w*16+16) x all 64 cols (4 accumulators).
//   Staging: async VMEM->LDS b128 copies into the spare buffer, issued before
//   the current tile's 32 WMMAs; one s_wait_asynccnt + barrier per K-tile.
//   A tile in LDS: [m][k], row stride 36 (b64 frag reads, conflict-free).
//   B tile in LDS: TRANS_B ? [n][k] stride 36 : [k][n] stride 68 (natural
//   layouts so the async copy needs no transpose).
// WMMA f32 16x16x4 fragment layout (wave32), lane = 16*g + r:
//   A frag: A[r][2g], A[r][2g+1]      B frag: B[2g][r], B[2g+1][r]
//   C/D: vgpr j -> C[j + 8*g][r]
// ---------------------------------------------------------------------------
template <bool TRANS_B, bool ACCUM, bool HAS_BIAS>
__global__ __launch_bounds__(256) void wmma_gemm_tiled(
    const float* __restrict__ A, int lda, long long strideA,
    const float* __restrict__ Bm, int ldb, long long strideB,
    const float* __restrict__ bias,
    float* __restrict__ C, int ldc, long long strideC,
    int K)
{
  constexpr int BM = 128, BN = 64, KT = 32;
  constexpr int ASTR = 36;                       // A row stride (pad)
  constexpr int BSTR = TRANS_B ? 36 : 68;        // B row stride (pad)
  constexpr int BELT = TRANS_B ? BN * BSTR : KT * BSTR;
  __shared__ float As[2][BM * ASTR];
  __shared__ float Bs[2][BELT];

  const int tid  = threadIdx.x;
  const int lane = tid & 31;
  const int wave = tid >> 5;
  const int g = lane >> 4;   // K-pair group
  const int r = lane & 15;   // A-row / B-col within fragment
  const int m0 = blockIdx.x * BM;
  const int n0 = blockIdx.y * BN;

  const float* Ab = A  + (long long)blockIdx.z * strideA;
  const float* Bb = Bm + (long long)blockIdx.z * strideB;
  float*       Cb = C  + (long long)blockIdx.z * strideC;

  // Issue the full K-tile copy (A: 1024 quads, B: 512 quads; 6 per thread).
  auto stage = [&](int k0, int buf) {
#pragma unroll
    for (int i = 0; i < 4; ++i) {
      const int idx = i * 256 + tid;
      const int row = idx >> 3, c4 = idx & 7;
      async_b128(Ab + (size_t)(m0 + row) * lda + k0 + c4 * 4,
                 &As[buf][row * ASTR + c4 * 4]);
    }
#pragma unroll
    for (int i = 0; i < 2; ++i) {
      const int idx = i * 256 + tid;
      if (TRANS_B) {   // global B is (N x K): natural [n][k]
        const int n = idx >> 3, c4 = idx & 7;
        async_b128(Bb + (size_t)(n0 + n) * ldb + k0 + c4 * 4,
                   &Bs[buf][n * BSTR + c4 * 4]);
      } else {         // global B is (K x N): natural [k][n]
        const int kk = idx >> 4, c4 = idx & 15;
        async_b128(Bb + (size_t)(k0 + kk) * ldb + n0 + c4 * 4,
                   &Bs[buf][kk * BSTR + c4 * 4]);
      }
    }
  };

  v8f acc[4] = {};

  stage(0, 0);
  wait_async0();
  __syncthreads();

  const int nk = K / KT;
  for (int kt = 0; kt < nk; ++kt) {
    const int cur = kt & 1;
    if (kt + 1 < nk) stage((kt + 1) * KT, 1 - cur);  // drains during WMMAs

    const float* as = &As[cur][(wave * 16 + r) * ASTR + 2 * g];
#pragma unroll
    for (int ks = 0; ks < 8; ++ks) {
      const v2f a = *(const v2f*)(as + ks * 4);
#pragma unroll
      for (int t = 0; t < 4; ++t) {
        v2f b;
        if (TRANS_B) {
          b = *(const v2f*)&Bs[cur][(t * 16 + r) * BSTR + ks * 4 + 2 * g];
        } else {
          b.x = Bs[cur][(ks * 4 + 2 * g) * BSTR + t * 16 + r];
          b.y = Bs[cur][(ks * 4 + 2 * g + 1) * BSTR + t * 16 + r];
        }
        acc[t] = __builtin_amdgcn_wmma_f32_16x16x4_f32(
            false, a, false, b, (short)0, acc[t], false, false);
      }
    }
    wait_async0();      // next tile fully resident in LDS
    __syncthreads();    // and all waves done reading buf[cur]
  }

  const int mw = m0 + wave * 16;
#pragma unroll
  for (int t = 0; t < 4; ++t) {
    const int n = n0 + t * 16 + r;
    const float badd = HAS_BIAS ? bias[n] : 0.0f;
#pragma unroll
    for (int j = 0; j < 8; ++j) {
      const size_t idx = (size_t)(mw + j + 8 * g) * ldc + n;
      float v = acc[t][j] + badd;
      if (ACCUM) v += Cb[idx];
      Cb[idx] = v;
    }
  }
}

// ---------------------------------------------------------------------------
// Direct (non-tiled) WMMA GEMM for the tiny N=16 state projections.
// One 16x16 tile per wave, 8 waves per block stacked in M.
// ---------------------------------------------------------------------------
__global__ __launch_bounds__(256) void wmma_gemm_small(
    const float* __restrict__ A, int lda,
    const float* __restrict__ Bm, int ldb,
    const float* __restrict__ bias,
    float* __restrict__ C, int ldc,
    int K)
{
  const int lane = threadIdx.x & 31;
  const int wave = threadIdx.x >> 5;
  const int g = lane >> 4;
  const int r = lane & 15;
  const int m0 = (blockIdx.x * 8 + wave) * 16;

  const float* Ap = A + (size_t)(m0 + r) * lda + 2 * g;
  v8f acc = {};

  for (int k = 0; k < K; k += 4) {
    v2f a;
    a.x = Ap[k];
    a.y = Ap[k + 1];
    v2f b;
    const float* Bp = Bm + (size_t)(k + 2 * g) * ldb + r;
    b.x = Bp[0];
    b.y = Bp[ldb];
    acc = __builtin_amdgcn_wmma_f32_16x16x4_f32(
        false, a, false, b, (short)0, acc, false, false);
  }

  const float badd = bias[r];
#pragma unroll
  for (int j = 0; j < 8; ++j)
    C[(size_t)(m0 + j + 8 * g) * ldc + r] = acc[j] + badd;
}

// ---------------------------------------------------------------------------
// Row softmax with pre-scale, in place. One 256-thread block per row.
// ---------------------------------------------------------------------------
__global__ __launch_bounds__(256) void softmax_rows(float* __restrict__ S,
                                                    int ncols, float scale)
{
  float* p = S + (size_t)blockIdx.x * ncols;
  const int tid = threadIdx.x, lane = tid & 31, wv = tid >> 5;
  __shared__ float red[8];

  float mx = -3.0e38f;
  for (int c = tid; c < ncols; c += 256) mx = fmaxf(mx, p[c] * scale);
  for (int off = 16; off; off >>= 1) mx = fmaxf(mx, __shfl_down(mx, off, 32));
  if (lane == 0) red[wv] = mx;
  __syncthreads();
  if (tid == 0) {
    float v = red[0];
#pragma unroll
    for (int i = 1; i < 8; ++i) v = fmaxf(v, red[i]);
    red[0] = v;
  }
  __syncthreads();
  mx = red[0];
  __syncthreads();

  float sum = 0.0f;
  for (int c = tid; c < ncols; c += 256) {
    const float e = __expf(p[c] * scale - mx);
    p[c] = e;
    sum += e;
  }
  for (int off = 16; off; off >>= 1) sum += __shfl_down(sum, off, 32);
  if (lane == 0) red[wv] = sum;
  __syncthreads();
  if (tid == 0) {
    float v = 0.0f;
#pragma unroll
    for (int i = 0; i < 8; ++i) v += red[i];
    red[0] = v;
  }
  __syncthreads();
  const float inv = 1.0f / red[0];
  for (int c = tid; c < ncols; c += 256) p[c] *= inv;
}

// ---------------------------------------------------------------------------
// Selective scan. grid = B * (D/128), block = 128 threads (one channel d each).
// h[n] = exp(softplus(dt)*A[d,n]) * h[n] + softplus(dt)*x*B[t,n]
// y    = sum_n h[n]*C[t,n]
// ---------------------------------------------------------------------------
__global__ __launch_bounds__(128) void selective_scan(
    const float* __restrict__ X, int ldx,          // (B,L,*) x at row stride ldx
    const float* __restrict__ DT,                  // (B,L,D)
    const float* __restrict__ A_log,               // (D,N)
    const float* __restrict__ Bm,                  // (B,L,N)
    const float* __restrict__ Cm,                  // (B,L,N)
    float* __restrict__ Y)                         // (B,L,D)
{
  const int dblocks = kD / 128;
  const int b = blockIdx.x / dblocks;
  const int d = (blockIdx.x % dblocks) * 128 + threadIdx.x;

  float Ar[kN];
#pragma unroll
  for (int n = 0; n < kN; ++n) Ar[n] = -__expf(A_log[d * kN + n]);

  float h[kN] = {};
  __shared__ float sB[kN];
  __shared__ float sC[kN];

  const float* xp = X + (size_t)b * kL * ldx + d;
  const float* dtp = DT + (size_t)b * kL * kD + d;
  const float* bp = Bm + (size_t)b * kL * kN;
  const float* cp = Cm + (size_t)b * kL * kN;
  float* yp = Y + (size_t)b * kL * kD + d;

  for (int t = 0; t < kL; ++t) {
    if (threadIdx.x < kN)
      sB[threadIdx.x] = bp[t * kN + threadIdx.x];
    else if (threadIdx.x < 2 * kN)
      sC[threadIdx.x - kN] = cp[t * kN + threadIdx.x - kN];
    __syncthreads();

    if (t + 1 < kL) {  // hide next step's latency (global_prefetch_b8)
      __builtin_prefetch(dtp + (size_t)(t + 1) * kD, 0, 1);
      __builtin_prefetch(xp + (size_t)(t + 1) * ldx, 0, 1);
    }

    const float dt = dtp[(size_t)t * kD];
    const float x = xp[(size_t)t * ldx];
    const float sp = (dt > 20.0f) ? dt : log1pf(__expf(dt));
    const float sx = sp * x;
    float y = 0.0f;
#pragma unroll
    for (int n = 0; n < kN; ++n) {
      const float dA = __expf(sp * Ar[n]);
      h[n] = dA * h[n] + sx * sB[n];
      y += h[n] * sC[n];
    }
    yp[(size_t)t * kD] = y;
    __syncthreads();
  }
}

// ---------------------------------------------------------------------------
// LayerNorm in place over rows of length D=512. Block = 128 threads.
// ---------------------------------------------------------------------------
__global__ __launch_bounds__(128) void layernorm_rows(
    float* __restrict__ Yio, const float* __restrict__ gamma,
    const float* __restrict__ beta)
{
  float* p = Yio + (size_t)blockIdx.x * kD;
  const int tid = threadIdx.x, lane = tid & 31, wv = tid >> 5;
  __shared__ float red[4];

  float v[4];
  float s = 0.0f;
#pragma unroll
  for (int i = 0; i < 4; ++i) { v[i] = p[tid + i * 128]; s += v[i]; }
  for (int off = 16; off; off >>= 1) s += __shfl_down(s, off, 32);
  if (lane == 0) red[wv] = s;
  __syncthreads();
  const float mu = (red[0] + red[1] + red[2] + red[3]) * (1.0f / kD);
  __syncthreads();

  float var = 0.0f;
#pragma unroll
  for (int i = 0; i < 4; ++i) { const float d = v[i] - mu; var += d * d; }
  for (int off = 16; off; off >>= 1) var += __shfl_down(var, off, 32);
  if (lane == 0) red[wv] = var;
  __syncthreads();
  var = (red[0] + red[1] + red[2] + red[3]) * (1.0f / kD);
  const float inv = rsqrtf(var + 1e-5f);
#pragma unroll
  for (int i = 0; i < 4; ++i) {
    const int c = tid + i * 128;
    p[c] = (v[i] - mu) * inv * gamma[c] + beta[c];
  }
}

// ---------------------------------------------------------------------------
// Host-side orchestration.
// ---------------------------------------------------------------------------
extern "C" void kernel_launch(void* const* d_in, const int* in_sizes, int n_in,
                              void* d_out, int out_size, void* d_ws, size_t ws_size,
                              hipStream_t stream) {
  (void)in_sizes; (void)n_in; (void)out_size; (void)ws_size;

  const float* text  = (const float*)d_in[0];   // (4,1024,512)
  const float* image = (const float*)d_in[1];
  const float* W_t2i = (const float*)d_in[2];   const float* b_t2i = (const float*)d_in[3];
  const float* W_i2t = (const float*)d_in[4];   const float* b_i2t = (const float*)d_in[5];
  const float* W_dt_t = (const float*)d_in[6];  const float* b_dt_t = (const float*)d_in[7];
  const float* W_dt_i = (const float*)d_in[8];  const float* b_dt_i = (const float*)d_in[9];
  const float* W_B_t = (const float*)d_in[10];  const float* b_B_t = (const float*)d_in[11];
  const float* W_B_i = (const float*)d_in[12];  const float* b_B_i = (const float*)d_in[13];
  const float* W_C_t = (const float*)d_in[14];  const float* b_C_t = (const float*)d_in[15];
  const float* W_C_i = (const float*)d_in[16];  const float* b_C_i = (const float*)d_in[17];
  const float* A_log = (const float*)d_in[18];
  const float* W_out = (const float*)d_in[19];  const float* b_out = (const float*)d_in[20];
  const float* gamma = (const float*)d_in[21];  const float* beta  = (const float*)d_in[22];

  float* ws = (float*)d_ws;
  // Workspace layout (float offsets)
  float* t2i    = ws;                       // 4M  (B,L,2D) interleaved [q|k]
  float* i2t    = ws + 4194304;             // 4M
  float* dt_t   = ws + 8388608;             // 2M
  float* dt_i   = ws + 10485760;            // 2M
  float* scores = ws + 12582912;            // 4M  (reused per direction)
  float* bfeat  = ws + 16777216;            // 2M  (reused per direction)
  float* bm_t   = ws + 18874368;            // 64K
  float* cm_t   = ws + 18939904;            // 64K
  float* bm_i   = ws + 19005440;            // 64K
  float* cm_i   = ws + 19070976;            // 64K
  float* scan_t = ws + 19136512;            // 2M
  float* scan_i = ws + 21233664;            // 2M

  float* out_text  = (float*)d_out;             // (4,1024,512)
  float* out_image = (float*)d_out + 2097152;

  const int M = kB * kL;                    // 4096
  const float scale = 0.04419417382415922f; // 1/sqrt(512)
  const long long sQK = (long long)kL * 2 * kD;   // per-batch stride in t2i/i2t
  const long long sSC = (long long)kL * kL;       // per-batch stride in scores
  const long long sBF = (long long)kL * kD;       // per-batch stride in bfeat

  // 1-2) input projections: t2i / i2t = X @ W + b   (M=4096,N=1024,K=512)
  wmma_gemm_tiled<false,false,true><<<dim3(32,16,1),256,0,stream>>>(
      text, kD, 0, W_t2i, 2*kD, 0, b_t2i, t2i, 2*kD, 0, kD);
  wmma_gemm_tiled<false,false,true><<<dim3(32,16,1),256,0,stream>>>(
      image, kD, 0, W_i2t, 2*kD, 0, b_i2t, i2t, 2*kD, 0, kD);

  // 3-4) dt projections from q halves (lda = 2D, offset 0)
  wmma_gemm_tiled<false,false,true><<<dim3(32,8,1),256,0,stream>>>(
      t2i, 2*kD, 0, W_dt_t, kD, 0, b_dt_t, dt_t, kD, 0, kD);
  wmma_gemm_tiled<false,false,true><<<dim3(32,8,1),256,0,stream>>>(
      i2t, 2*kD, 0, W_dt_i, kD, 0, b_dt_i, dt_i, kD, 0, kD);

  // ---------------- direction t2i: q_text vs k_image ----------------
  // scores[b] = q_t2i @ k_i2t^T  (1024x1024, K=512, batched over B)
  wmma_gemm_tiled<true,false,false><<<dim3(8,16,kB),256,0,stream>>>(
      t2i, 2*kD, sQK, i2t + kD, 2*kD, sQK, nullptr, scores, kL, sSC, kD);
  softmax_rows<<<dim3(kB*kL),256,0,stream>>>(scores, kL, scale);
  // bfeat[b] = P @ k_i2t  (1024x512, K=1024)
  wmma_gemm_tiled<false,false,false><<<dim3(8,8,kB),256,0,stream>>>(
      scores, kL, sSC, i2t + kD, 2*kD, sQK, nullptr, bfeat, kD, sBF, kL);
  // B/C projections (N=16)
  wmma_gemm_small<<<dim3(32),256,0,stream>>>(bfeat, kD, W_B_t, kN, b_B_t, bm_t, kN, kD);
  wmma_gemm_small<<<dim3(32),256,0,stream>>>(bfeat, kD, W_C_t, kN, b_C_t, cm_t, kN, kD);
  // scan
  selective_scan<<<dim3(kB*(kD/128)),128,0,stream>>>(
      t2i, 2*kD, dt_t, A_log, bm_t, cm_t, scan_t);

  // ---------------- direction i2t: q_image vs k_text ----------------
  wmma_gemm_tiled<true,false,false><<<dim3(8,16,kB),256,0,stream>>>(
      i2t, 2*kD, sQK, t2i + kD, 2*kD, sQK, nullptr, scores, kL, sSC, kD);
  softmax_rows<<<dim3(kB*kL),256,0,stream>>>(scores, kL, scale);
  wmma_gemm_tiled<false,false,false><<<dim3(8,8,kB),256,0,stream>>>(
      scores, kL, sSC, t2i + kD, 2*kD, sQK, nullptr, bfeat, kD, sBF, kL);
  wmma_gemm_small<<<dim3(32),256,0,stream>>>(bfeat, kD, W_B_i, kN, b_B_i, bm_i, kN, kD);
  wmma_gemm_small<<<dim3(32),256,0,stream>>>(bfeat, kD, W_C_i, kN, b_C_i, cm_i, kN, kD);
  selective_scan<<<dim3(kB*(kD/128)),128,0,stream>>>(
      i2t, 2*kD, dt_i, A_log, bm_i, cm_i, scan_i);

  // ---------------- output projection: concat split into two GEMMs ----------
  // out = feats @ W_out[:512,:] + b_out ; out += scan @ W_out[512:,:]
  wmma_gemm_tiled<false,false,true><<<dim3(32,8,1),256,0,stream>>>(
      text, kD, 0, W_out, kD, 0, b_out, out_text, kD, 0, kD);
  wmma_gemm_tiled<false,true,false><<<dim3(32,8,1),256,0,stream>>>(
      scan_t, kD, 0, W_out + (size_t)kD * kD, kD, 0, nullptr, out_text, kD, 0, kD);
  wmma_gemm_tiled<false,false,true><<<dim3(32,8,1),256,0,stream>>>(
      image, kD, 0, W_out, kD, 0, b_out, out_image, kD, 0, kD);
  wmma_gemm_tiled<false,true,false><<<dim3(32,8,1),256,0,stream>>>(
      scan_i, kD, 0, W_out + (size_t)kD * kD, kD, 0, nullptr, out_image, kD, 0, kD);

  // ---------------- layernorm in place over both outputs --------------------
  layernorm_rows<<<dim3(2 * M),128,0,stream>>>((float*)d_out, gamma, beta);
}